// TemporalCrudeAttention_87093346828767
// MI455X (gfx1250) — compile-verified
//
#include <hip/hip_runtime.h>

typedef __attribute__((ext_vector_type(2))) float v2f;
typedef __attribute__((ext_vector_type(8))) float v8f;

#define TT 2048
#define DD 64
#define TC 32          // time rows per chunk
#define XP 68          // padded LDS row pitch (floats) -> conflict-free column reads

__device__ __forceinline__ float fast_exp(float v) {
  return __builtin_amdgcn_exp2f(v * 1.44269504088896340736f);
}
__device__ __forceinline__ float fast_tanh(float v) {
  // tanh(v) = 1 - 2/(exp(2v)+1); saturates correctly for |v| large
  float t = __builtin_amdgcn_exp2f(v * 2.88539008177792681472f);
  return 1.0f - 2.0f * __builtin_amdgcn_rcpf(t + 1.0f);
}

// Async direct global->LDS copy of 32 bytes (2 x b128), CDNA5 path (ASYNCcnt).
// INST_OFFSET is added to BOTH the LDS and the global address per the ISA.
__device__ __forceinline__ void async_copy32(unsigned lds_addr, const float* gptr) {
  asm volatile(
      "global_load_async_to_lds_b128 %0, %1, off\n\t"
      "global_load_async_to_lds_b128 %0, %1, off offset:16"
      :: "v"(lds_addr), "v"(gptr) : "memory");
}
__device__ __forceinline__ void wait_async0() {
  asm volatile("s_wait_asynccnt 0x0" ::: "memory");
}

__global__ __launch_bounds__(256, 1)
void temporal_attn_kernel(const float* __restrict__ x,
                          const float* __restrict__ W,
                          const float* __restrict__ bias,
                          float* __restrict__ out)
{
  __shared__ float lds_W[DD * DD];    // W[k][n], row-major
  __shared__ float lds_X[TC * XP];    // x chunk, padded
  __shared__ float lds_Bc[TC * XP];   // bias chunk, padded
  __shared__ float lds_O[TC * XP];    // output staging, padded
  __shared__ float s_part[DD * 4];    // deterministic 4-way partial sums per column
  __shared__ float s_rs[DD];          // 1/sum per column

  const int tid   = threadIdx.x;
  const int batch = blockIdx.x;
  const float* xb = x   + (size_t)batch * TT * DD;
  float*       ob = out + (size_t)batch * TT * DD;

  // ---- cooperative load of W (4096 floats) into LDS ----
  #pragma unroll
  for (int i = 0; i < 4; ++i) {
    int idx = i * 1024 + tid * 4;
    *(float4*)&lds_W[idx] = *(const float4*)&W[idx];
  }
  __syncthreads();

  const int wave   = tid >> 5;
  const int lane   = tid & 31;
  const int laneLo = lane & 15;
  const int hi     = lane >> 4;            // 0: lanes 0-15, 1: lanes 16-31
  const int r0     = (wave >> 2) << 4;     // row tile base within chunk (0 or 16)
  const int n0     = (wave & 3) << 4;      // col tile base (0,16,32,48)
  const int col    = n0 + laneLo;          // this lane's output column d

  // ---- preload B-operand (W columns n0..n0+15) into registers ----
  // B 4x16 f32 layout: V0 = {K=0 lanes0-15, K=2 lanes16-31}, V1 = {K=1, K=3}
  v2f breg[16];
  #pragma unroll
  for (int kk = 0; kk < 16; ++kk) {
    int k0 = kk * 4 + hi * 2;
    breg[kk].x = lds_W[(k0    ) * DD + col];
    breg[kk].y = lds_W[(k0 + 1) * DD + col];
  }

  const int ldr = tid >> 3;          // chunk-load row 0..31
  const int ldc = (tid & 7) * 8;     // chunk-load col 0,8,...,56

  // per-thread LDS destinations (low 32 bits of flat address == LDS offset)
  const unsigned lx = (unsigned)(uintptr_t)&lds_X [ldr * XP + ldc];
  const unsigned lb = (unsigned)(uintptr_t)&lds_Bc[ldr * XP + ldc];

  // ================= PASS 1: s[d] = sum_t exp(tanh(xW+b)) =================
  float spart = 0.0f;
  {
    const float* gxp = xb   + (size_t)ldr * DD + ldc;
    const float* gbp = bias + (size_t)ldr * DD + ldc;
    for (int t0 = 0; t0 < TT; t0 += TC) {
      __syncthreads();   // protect lds_X/lds_Bc from previous iteration readers
      async_copy32(lx, gxp);
      async_copy32(lb, gbp);
      wait_async0();
      gxp += TC * DD;
      gbp += TC * DD;
      __syncthreads();

      // A 16x4 f32 layout: V0 = {K=0 lanes0-15, K=2 lanes16-31}, V1 = {K=1, K=3}
      v8f acc = {0.f, 0.f, 0.f, 0.f, 0.f, 0.f, 0.f, 0.f};
      const int arow = r0 + laneLo;
      #pragma unroll
      for (int kk = 0; kk < 16; ++kk) {
        int k = kk * 4 + hi * 2;
        v2f a = *(const v2f*)&lds_X[arow * XP + k];
        acc = __builtin_amdgcn_wmma_f32_16x16x4_f32(false, a, false, breg[kk],
                                                    (short)0, acc, false, false);
      }
      // D layout: VGPR r -> row r (lanes 0-15) / row r+8 (lanes 16-31)
      #pragma unroll
      for (int r = 0; r < 8; ++r) {
        int row = r0 + r + hi * 8;
        float e = fast_tanh(acc[r] + lds_Bc[row * XP + col]);
        spart  += fast_exp(e);
      }
    }
  }
  // deterministic merge: 4 partials per column (2 row-waves x 2 half-waves)
  s_part[col * 4 + ((wave >> 2) * 2 + hi)] = spart;
  __syncthreads();
  if (tid < DD) {
    float s = s_part[tid * 4] + s_part[tid * 4 + 1] +
              s_part[tid * 4 + 2] + s_part[tid * 4 + 3];
    s_rs[tid] = __builtin_amdgcn_rcpf(s);
  }
  __syncthreads();

  // ================= PASS 2: out = x * exp(e) / s =================
  const float rs = s_rs[col];
  {
    const float* gxp = xb   + (size_t)ldr * DD + ldc;
    const float* gbp = bias + (size_t)ldr * DD + ldc;
    for (int t0 = 0; t0 < TT; t0 += TC) {
      __syncthreads();
      async_copy32(lx, gxp);
      async_copy32(lb, gbp);
      wait_async0();
      gxp += TC * DD;
      gbp += TC * DD;
      __syncthreads();

      v8f acc = {0.f, 0.f, 0.f, 0.f, 0.f, 0.f, 0.f, 0.f};
      const int arow = r0 + laneLo;
      #pragma unroll
      for (int kk = 0; kk < 16; ++kk) {
        int k = kk * 4 + hi * 2;
        v2f a = *(const v2f*)&lds_X[arow * XP + k];
        acc = __builtin_amdgcn_wmma_f32_16x16x4_f32(false, a, false, breg[kk],
                                                    (short)0, acc, false, false);
      }
      #pragma unroll
      for (int r = 0; r < 8; ++r) {
        int row  = r0 + r + hi * 8;
        float e  = fast_tanh(acc[r] + lds_Bc[row * XP + col]);
        float xv = lds_X[row * XP + col];
        lds_O[row * XP + col] = xv * fast_exp(e) * rs;
      }
      __syncthreads();
      // coalesced b128 stores from staging
      float4 o0 = *(const float4*)&lds_O[ldr * XP + ldc];
      float4 o1 = *(const float4*)&lds_O[ldr * XP + ldc + 4];
      *(float4*)(ob + (size_t)(t0 + ldr) * DD + ldc)     = o0;
      *(float4*)(ob + (size_t)(t0 + ldr) * DD + ldc + 4) = o1;
    }
  }
}

extern "C" void kernel_launch(void* const* d_in, const int* in_sizes, int n_in,
                              void* d_out, int out_size, void* d_ws, size_t ws_size,
                              hipStream_t stream) {
  const float* x    = (const float*)d_in[0];
  const float* W    = (const float*)d_in[1];
  const float* bias = (const float*)d_in[2];
  float* out        = (float*)d_out;
  int Bn = in_sizes[0] / (TT * DD);   // 256
  temporal_attn_kernel<<<dim3(Bn), dim3(256), 0, stream>>>(x, W, bias, out);
}